// Mamba_11931419148962
// MI455X (gfx1250) — compile-verified
//
#include <hip/hip_runtime.h>
#include <hip/hip_bf16.h>
#include <cstdint>

typedef __attribute__((ext_vector_type(2))) float v2f;
typedef __attribute__((ext_vector_type(8))) float v8f;

#define D_MODEL 1024
#define D_INNER 2048
#define D_STATE 16
#define DT_RANK 64
#define D_CONV  4
#define N_LAYER 2
#define VOCAB   50280
#define SEQ_L   1024

__device__ __forceinline__ float silu_f(float v) {
    return v / (1.0f + __expf(-v));
}
__device__ __forceinline__ float softplus_f(float v) {
    return (v > 20.0f) ? v : log1pf(__expf(v));
}

// ---------------------------------------------------------------------------
// Embedding gather: x[t,:] = E[ids[t],:]   (one block per token, float4 copy)
// ---------------------------------------------------------------------------
__global__ void __launch_bounds__(256)
embed_k(const int* __restrict__ ids, const float* __restrict__ E,
        float* __restrict__ x) {
    int t = blockIdx.x;
    int i = threadIdx.x;                     // 256 threads * float4 = 1024 f32
    int id = ids[t];
    ((float4*)(x + (size_t)t * D_MODEL))[i] =
        ((const float4*)(E + (size_t)id * D_MODEL))[i];
}

// ---------------------------------------------------------------------------
// RMSNorm over rows of 1024: o = x * rsqrt(mean(x^2)+eps) * scale
// ---------------------------------------------------------------------------
__global__ void __launch_bounds__(256)
rmsnorm_k(const float* __restrict__ x, const float* __restrict__ scale,
          float* __restrict__ o) {
    __shared__ float red[8];
    int t = blockIdx.x, i = threadIdx.x;
    float4 v = ((const float4*)(x + (size_t)t * D_MODEL))[i];
    float s = v.x * v.x + v.y * v.y + v.z * v.z + v.w * v.w;
#pragma unroll
    for (int off = 16; off > 0; off >>= 1) s += __shfl_down(s, off);
    if ((i & 31) == 0) red[i >> 5] = s;
    __syncthreads();
    float tot = red[0] + red[1] + red[2] + red[3] +
                red[4] + red[5] + red[6] + red[7];
    float rr = rsqrtf(tot * (1.0f / (float)D_MODEL) + 1e-5f);
    float4 sc = ((const float4*)scale)[i];
    float4 r;
    r.x = v.x * rr * sc.x;
    r.y = v.y * rr * sc.y;
    r.z = v.z * rr * sc.z;
    r.w = v.w * rr * sc.w;
    ((float4*)(o + (size_t)t * D_MODEL))[i] = r;
}

// ---------------------------------------------------------------------------
// WMMA f32 GEMM:  D[m,n] = act( sum_k A[m,k]*B(k,n) + bias[n] ) + Cadd[m,n]
//   B(k,n) = TRANSB ? B[n*ldb + k] : B[k*ldb + n]
// One wave per block computing MT*16 (M) x NT*16 (N) with
// v_wmma_f32_16x16x4_f32. B fragments are reused across MT M-tiles, A
// fragments across NT N-tiles. All addressing via 32-bit offsets from
// uniform bases (saddr + voffset + imm), so unrolled k-steps fold the
// address math into instruction immediates.
// M multiple of 16*MT, K multiple of 16. GUARD=false requires n in-bounds.
// ---------------------------------------------------------------------------
template<int MT, int NT, bool TRANSB, bool GUARD>
__global__ void __launch_bounds__(32)
gemm_wmma(const float* __restrict__ A, int lda,
          const float* __restrict__ B, int ldb,
          const float* __restrict__ bias, int act,
          const float* __restrict__ Cadd, int ldc,
          float* __restrict__ D, int ldd,
          int N, int K, int tileNbase) {
    const int lane   = threadIdx.x & 31;
    const int idx    = lane & 15;          // M index in tile (A,C,D) / N (B)
    const int half   = lane >> 4;          // K phase: 0 -> K+0/1, 1 -> K+2/3
    const int kph    = half * 2;
    const int tileM  = blockIdx.y * (16 * MT);
    const int tileN0 = tileNbase + blockIdx.x * (NT * 16);

    v8f c[MT][NT] = {};

    int aoff[MT];
#pragma unroll
    for (int i = 0; i < MT; ++i)
        aoff[i] = (tileM + i * 16 + idx) * lda + kph;

    int boff[NT];
#pragma unroll
    for (int j = 0; j < NT; ++j) {
        int n = tileN0 + j * 16 + idx;
        boff[j] = TRANSB ? (n * ldb + kph) : (kph * ldb + n);
    }

#pragma unroll 4
    for (int k = 0; k < K; k += 4) {
        v2f a[MT];
#pragma unroll
        for (int i = 0; i < MT; ++i)
            a[i] = *(const v2f*)(A + aoff[i] + k);
#pragma unroll
        for (int j = 0; j < NT; ++j) {
            v2f b;
            if (GUARD) {
                int n = tileN0 + j * 16 + idx;
                b = 0.0f;
                if (n < N) {
                    if (TRANSB) {
                        b = *(const v2f*)(B + boff[j] + k);
                    } else {
                        b.x = B[boff[j] + k * ldb];
                        b.y = B[boff[j] + k * ldb + ldb];
                    }
                }
            } else {
                if (TRANSB) {
                    b = *(const v2f*)(B + boff[j] + k);
                } else {
                    b.x = B[boff[j] + k * ldb];
                    b.y = B[boff[j] + k * ldb + ldb];
                }
            }
#pragma unroll
            for (int i = 0; i < MT; ++i)
                c[i][j] = __builtin_amdgcn_wmma_f32_16x16x4_f32(
                    false, a[i], false, b, (short)0, c[i][j], false, false);
        }
    }

#pragma unroll
    for (int i = 0; i < MT; ++i) {
#pragma unroll
        for (int j = 0; j < NT; ++j) {
            int n = tileN0 + j * 16 + idx;
            if (GUARD && n >= N) continue;
            float bv = bias ? bias[n] : 0.0f;
            int mbase = tileM + i * 16 + half * 8;     // VGPR r holds M=mbase+r
            int doff = mbase * ldd + n;
            int coff = mbase * ldc + n;
#pragma unroll
            for (int r = 0; r < 8; ++r) {
                float v = c[i][j][r] + bv;
                if (act == 1) v = softplus_f(v);
                if (Cadd) v += Cadd[coff + r * ldc];
                D[doff + r * ldd] = v;
            }
        }
    }
}

// ---------------------------------------------------------------------------
// Depthwise causal conv (W=4) + bias + SiLU.
// Input = xr[:, 0:2048] (row stride 4096), output xc (row stride 2048).
// ---------------------------------------------------------------------------
__global__ void __launch_bounds__(256)
conv_silu_k(const float* __restrict__ xr, const float* __restrict__ cw,
            const float* __restrict__ cb, float* __restrict__ xc) {
    int i = blockIdx.x * 256 + threadIdx.x;     // over SEQ_L * D_INNER
    int t = i >> 11;
    int d = i & (D_INNER - 1);
    float acc = cb[d];
#pragma unroll
    for (int w = 0; w < D_CONV; ++w) {
        int tt = t - (D_CONV - 1) + w;
        if (tt >= 0)
            acc += cw[w * D_INNER + d] * xr[(size_t)tt * (2 * D_INNER) + d];
    }
    xc[i] = silu_f(acc);
}

// ---------------------------------------------------------------------------
// Selective scan, one thread per channel d; h[16],A[16] in registers.
// y[t,d] = (sum_n h_n * C[t,n] + u*Dp) * silu(res[t,d])
// ---------------------------------------------------------------------------
__global__ void __launch_bounds__(256)
scan_k(const float* __restrict__ delta, const float* __restrict__ xc,
       const float* __restrict__ xdbl, const float* __restrict__ A_log,
       const float* __restrict__ Dskip, const float* __restrict__ xr,
       float* __restrict__ y) {
    int d = blockIdx.x * 256 + threadIdx.x;
    float A[D_STATE], h[D_STATE];
#pragma unroll
    for (int n = 0; n < D_STATE; ++n) {
        A[n] = -__expf(A_log[(size_t)d * D_STATE + n]);
        h[n] = 0.0f;
    }
    float Dp = Dskip[d];
    for (int t = 0; t < SEQ_L; ++t) {
        float dl = delta[(size_t)t * D_INNER + d];
        float u  = xc[(size_t)t * D_INNER + d];
        const float* Bp = xdbl + (size_t)t * (DT_RANK + 2 * D_STATE) + DT_RANK;
        const float* Cp = Bp + D_STATE;
        float acc = 0.0f;
#pragma unroll
        for (int q = 0; q < 4; ++q) {
            float4 B4 = ((const float4*)Bp)[q];
            float4 C4 = ((const float4*)Cp)[q];
            float bb[4] = {B4.x, B4.y, B4.z, B4.w};
            float cc[4] = {C4.x, C4.y, C4.z, C4.w};
#pragma unroll
            for (int e = 0; e < 4; ++e) {
                int n = q * 4 + e;
                float dA = __expf(dl * A[n]);
                h[n] = dA * h[n] + dl * bb[e] * u;
                acc += h[n] * cc[e];
            }
        }
        float yv = acc + u * Dp;
        float r = xr[(size_t)t * (2 * D_INNER) + D_INNER + d];  // res half
        y[(size_t)t * D_INNER + d] = yv * silu_f(r);
    }
}

// ---------------------------------------------------------------------------
// Host orchestration
// ---------------------------------------------------------------------------
template<int MT, int NT, bool TRANSB, bool GUARD>
static inline void run_gemm(const float* A, int lda, const float* B, int ldb,
                            const float* bias, int act,
                            const float* Cadd, int ldc, float* D, int ldd,
                            int M, int N, int K, int nblk, int tileNbase,
                            hipStream_t stream) {
    dim3 grid(nblk, M / (16 * MT));
    gemm_wmma<MT, NT, TRANSB, GUARD><<<grid, 32, 0, stream>>>(
        A, lda, B, ldb, bias, act, Cadd, ldc, D, ldd, N, K, tileNbase);
}

extern "C" void kernel_launch(void* const* d_in, const int* in_sizes, int n_in,
                              void* d_out, int out_size, void* d_ws, size_t ws_size,
                              hipStream_t stream) {
    const int*   ids      = (const int*)d_in[0];
    const float* E        = (const float*)d_in[1];
    const float* norm_s   = (const float*)d_in[2];
    const float* in_w     = (const float*)d_in[3];
    const float* conv_w   = (const float*)d_in[4];
    const float* conv_b   = (const float*)d_in[5];
    const float* xp_w     = (const float*)d_in[6];
    const float* dt_w     = (const float*)d_in[7];
    const float* dt_b     = (const float*)d_in[8];
    const float* A_log    = (const float*)d_in[9];
    const float* D_skip   = (const float*)d_in[10];
    const float* out_w    = (const float*)d_in[11];
    const float* norm_f   = (const float*)d_in[12];
    float*       logits   = (float*)d_out;

    // workspace layout (floats)
    float* ws    = (float*)d_ws;
    float* x     = ws;                         // 1M   (1024x1024)
    float* x2    = x     + (1u << 20);         // 1M
    float* xn    = x2    + (1u << 20);         // 1M
    float* xr    = xn    + (1u << 20);         // 4M   (1024x4096)
    float* xc    = xr    + (1u << 22);         // 2M   (1024x2048)
    float* xdbl  = xc    + (1u << 21);         // 96K  (1024x96) -> reserve 128K
    float* delta = xdbl  + (1u << 17);         // 2M   (1024x2048)
    float* yb    = delta + (1u << 21);         // 2M   (1024x2048)
    (void)in_sizes; (void)n_in; (void)out_size; (void)ws_size;

    // 1. embedding gather
    embed_k<<<SEQ_L, 256, 0, stream>>>(ids, E, x);

    const int XD = DT_RANK + 2 * D_STATE;      // 96

    for (int l = 0; l < N_LAYER; ++l) {
        float* cur = (l == 0) ? x : x2;
        float* nxt = (l == 0) ? x2 : x;

        // rmsnorm
        rmsnorm_k<<<SEQ_L, 256, 0, stream>>>(cur, norm_s + (size_t)l * D_MODEL, xn);

        // xr = xn @ in_proj_w[l]   (1024x1024 @ 1024x4096), 64 x-blocks
        run_gemm<2, 4, false, false>(xn, D_MODEL,
                                     in_w + (size_t)l * D_MODEL * 2 * D_INNER,
                                     2 * D_INNER, nullptr, 0, nullptr, 0,
                                     xr, 2 * D_INNER,
                                     SEQ_L, 2 * D_INNER, D_MODEL, 64, 0, stream);

        // depthwise conv + bias + silu on first half of xr -> xc
        conv_silu_k<<<(SEQ_L * D_INNER) / 256, 256, 0, stream>>>(
            xr, conv_w + (size_t)l * D_CONV * D_INNER,
            conv_b + (size_t)l * D_INNER, xc);

        // x_dbl = xc @ x_proj_w[l]   (1024x2048 @ 2048x96): NT=2 -> 3 blocks
        run_gemm<2, 2, false, false>(xc, D_INNER,
                                     xp_w + (size_t)l * D_INNER * XD, XD,
                                     nullptr, 0, nullptr, 0, xdbl, XD,
                                     SEQ_L, XD, D_INNER, 3, 0, stream);

        // delta = softplus(dt @ dt_proj_w[l] + dt_b[l])  (1024x64 @ 64x2048)
        run_gemm<2, 4, false, false>(xdbl, XD,
                                     dt_w + (size_t)l * DT_RANK * D_INNER, D_INNER,
                                     dt_b + (size_t)l * D_INNER, /*act=*/1,
                                     nullptr, 0, delta, D_INNER,
                                     SEQ_L, D_INNER, DT_RANK, 32, 0, stream);

        // selective scan (+ u*D, * silu(res)) -> yb
        scan_k<<<D_INNER / 256, 256, 0, stream>>>(
            delta, xc, xdbl,
            A_log + (size_t)l * D_INNER * D_STATE,
            D_skip + (size_t)l * D_INNER, xr, yb);

        // nxt = yb @ out_proj_w[l] + cur  (residual fused via Cadd)
        run_gemm<2, 4, false, false>(yb, D_INNER,
                                     out_w + (size_t)l * D_INNER * D_MODEL, D_MODEL,
                                     nullptr, 0, cur, D_MODEL, nxt, D_MODEL,
                                     SEQ_L, D_MODEL, D_INNER, 16, 0, stream);
    }

    // final rmsnorm (result of layer 1 is back in x)
    rmsnorm_k<<<SEQ_L, 256, 0, stream>>>(x, norm_f, xn);

    // logits = xn @ E^T   (1024x1024 @ 1024x50280, B transposed)
    // main: 785 blocks * 64 cols = 50240, unguarded
    run_gemm<2, 4, true, false>(xn, D_MODEL, E, D_MODEL,
                                nullptr, 0, nullptr, 0, logits, VOCAB,
                                SEQ_L, VOCAB, D_MODEL, 785, 0, stream);
    // tail: ragged last 40 columns, guarded
    run_gemm<2, 4, true, true>(xn, D_MODEL, E, D_MODEL,
                               nullptr, 0, nullptr, 0, logits, VOCAB,
                               SEQ_L, VOCAB, D_MODEL, 1, 50240, stream);
}